// GridPromptGenerator_82162724373016
// MI455X (gfx1250) — compile-verified
//
#include <hip/hip_runtime.h>
#include <stdint.h>

#define NUM_GRID  16
#define THRESHOLD 0.65f
#define NUM_FG    40
#define MAX_PTS   42
#define T_ 8
#define C_ 4
#define H_ 1024
#define W_ 1024

typedef unsigned long long u64;
typedef unsigned int u32;

__device__ __forceinline__ u64 umax64(u64 a, u64 b) { return a > b ? a : b; }
__device__ __forceinline__ u64 umin64(u64 a, u64 b) { return a < b ? a : b; }

__device__ __forceinline__ void async_copy_b128(unsigned lds_off, const void* gp) {
    // CDNA5 async global->LDS copy, tracked by ASYNCcnt (no VGPR return data)
    asm volatile("global_load_async_to_lds_b128 %0, %1, off"
                 :: "v"(lds_off), "v"(gp) : "memory");
}

__device__ __forceinline__ void wait_async0() {
#if __has_builtin(__builtin_amdgcn_s_wait_asynccnt)
    __builtin_amdgcn_s_wait_asynccnt(0);
#else
    asm volatile("s_wait_asynccnt 0" ::: "memory");
#endif
}

// One workgroup per (map, cell): async-stage a 64x64 f32 tile into LDS, then
// reduce to 2 packed u64 keys: {unconditional argmax, unconditional argmin}.
// (fg/thresholded cell max is derived later: it equals the unconditional max
//  whenever that max exceeds THRESHOLD, with identical first-index tie-break.)
__global__ __launch_bounds__(256) void cell_reduce_kernel(const float* __restrict__ sim,
                                                          u64* __restrict__ ws) {
    __shared__ __align__(16) float tile[64 * 64];
    __shared__ u64 smx[256], smn[256];

    const int tid  = threadIdx.x;
    const int g    = blockIdx.x;          // 0..8191
    const int map  = g >> 8;              // t*C + c
    const int cell = g & 255;             // cy*16 + cx
    const int row0 = (cell >> 4) << 6;
    const int col0 = (cell & 15) << 6;
    const float* base = sim + ((size_t)map << 20) + (size_t)row0 * W_ + col0;

    // Stage the whole tile (16 KB) with async b128 copies: 4 per thread.
#pragma unroll
    for (int k = 0; k < 4; ++k) {
        const int j  = tid + (k << 8);        // 0..1023 float4 slots
        const int r  = j >> 4;
        const int c4 = (j & 15) << 2;
        const float* gp = base + (size_t)r * W_ + c4;
        const unsigned loff = (unsigned)(uintptr_t)(&tile[j << 2]);
        async_copy_b128(loff, gp);
    }
    wait_async0();  // own copies complete -> own LDS slots valid

    // Hot loop: pure 32-bit ops. Elements are visited in strictly increasing
    // flat-index order per thread, so strict compares give exact
    // first-occurrence argmax/argmin tie-breaks.
    float vmax = -1.0f; u32 imax = 0;
    float vmin =  2.0f; u32 imin = 0;
#pragma unroll
    for (int k = 0; k < 4; ++k) {
        const int j  = tid + (k << 8);
        const int r  = j >> 4;
        const int c4 = (j & 15) << 2;
        const float4 v = *(const float4*)(&tile[j << 2]);
        const u32 idx0 = (u32)((row0 + r) * W_ + col0 + c4);
        const float vv[4] = {v.x, v.y, v.z, v.w};
#pragma unroll
        for (int e = 0; e < 4; ++e) {
            const u32 idx = idx0 | (u32)e;           // idx0 is 4-aligned
            const bool gt = vv[e] > vmax;
            vmax = gt ? vv[e] : vmax;
            imax = gt ? idx   : imax;
            const bool lt = vv[e] < vmin;
            vmin = lt ? vv[e] : vmin;
            imin = lt ? idx   : imin;
        }
    }

    // Pack once per thread for the cross-thread reduce (monotone for v>=0):
    // bigger key = bigger value, ties -> smaller flat index.
    u64 mxk = ((u64)__float_as_uint(vmax) << 32) | (u64)(0xFFFFFFFFu - imax);
    u64 mnk = ((u64)__float_as_uint(vmin) << 32) | (u64)imin;

    smx[tid] = mxk;
    smn[tid] = mnk;
    __syncthreads();
    for (int s = 128; s > 0; s >>= 1) {
        if (tid < s) {
            smx[tid] = umax64(smx[tid], smx[tid + s]);
            smn[tid] = umin64(smn[tid], smn[tid + s]);
        }
        __syncthreads();
    }
    if (tid == 0) {
        u64* o = ws + (size_t)g * 2;
        o[0] = smx[0];
        o[1] = smn[0];
    }
}

// One workgroup per map: rank cells, pick top-40 fg, global max fallback,
// global min as bg point; emit 42x4 rows + count.
__global__ __launch_bounds__(256) void select_kernel(const u64* __restrict__ ws,
                                                     const int* __restrict__ ori,
                                                     float* __restrict__ out_pts,
                                                     int* __restrict__ out_num) {
    __shared__ u64 rkey[256], smx[256], smn[256];
    __shared__ float fx[NUM_FG], fy[NUM_FG], fs[NUM_FG];

    const int tid = threadIdx.x;       // == cell id
    const int map = blockIdx.x;        // t*C + c
    const int t   = map / C_;

    const u64* cb = ws + (size_t)map * 256 * 2;
    const u64 mxk = cb[(size_t)tid * 2 + 0];
    const u64 mnk = cb[(size_t)tid * 2 + 1];

    // Derive the thresholded (foreground) cell key from the unconditional max.
    const float cellmax = __uint_as_float((u32)(mxk >> 32));
    const u64 fgk = (cellmax > THRESHOLD) ? mxk : 0;

    smx[tid] = mxk;
    smn[tid] = mnk;
    // rank key: score bits then lower-cell-id-first tie break (stable argsort)
    const u64 rk = (fgk == 0) ? 0 : (((fgk >> 32) << 8) | (u64)(255 - tid));
    rkey[tid] = rk;
    if (tid < NUM_FG) { fx[tid] = 0.f; fy[tid] = 0.f; fs[tid] = 0.f; }

    const int valid  = (fgk != 0);
    const int vcount = __syncthreads_count(valid);   // barrier + popcount

    for (int s = 128; s > 0; s >>= 1) {
        if (tid < s) {
            smx[tid] = umax64(smx[tid], smx[tid + s]);
            smn[tid] = umin64(smn[tid], smn[tid + s]);
        }
        __syncthreads();
    }

    int rank = 0;
    for (int j = 0; j < 256; ++j) rank += (rkey[j] > rk) ? 1 : 0;

    if (valid && rank < NUM_FG) {
        const u32 idx = 0xFFFFFFFFu - (u32)(fgk & 0xFFFFFFFFu);
        fx[rank] = (float)(idx & (W_ - 1));
        fy[rank] = (float)(idx >> 10);
        fs[rank] = __uint_as_float((u32)(fgk >> 32));
    }

    const int anyfg   = (vcount > 0);
    const int nvalid  = (vcount < NUM_FG) ? vcount : NUM_FG;
    const int fgcount = anyfg ? nvalid : 1;
    __syncthreads();

    if (!anyfg && tid == 0) {   // global-argmax fallback into slot 0
        const u64 gk  = smx[0];
        const u32 idx = 0xFFFFFFFFu - (u32)(gk & 0xFFFFFFFFu);
        fx[0] = (float)(idx & (W_ - 1));
        fy[0] = (float)(idx >> 10);
        fs[0] = __uint_as_float((u32)(gk >> 32));
    }
    __syncthreads();

    const float sy = (float)ori[2 * t]     / (float)H_;
    const float sx = (float)ori[2 * t + 1] / (float)W_;

    if (tid < MAX_PTS) {
        float4 row = make_float4(0.f, 0.f, 0.f, 0.f);
        if (tid == fgcount) {               // background point = global min
            const u64 gk  = smn[0];
            const u32 idx = (u32)(gk & 0xFFFFFFFFu);
            row.x = (float)(idx & (W_ - 1)) * sx;
            row.y = (float)(idx >> 10) * sy;
            row.z = __uint_as_float((u32)(gk >> 32));
            row.w = 0.f;
        } else if (tid < fgcount) {         // foreground points
            row.x = fx[tid] * sx;
            row.y = fy[tid] * sy;
            row.z = fs[tid];
            row.w = 1.f;
        }
        ((float4*)out_pts)[(size_t)map * MAX_PTS + tid] = row;
    }
    if (tid == 0) out_num[map] = fgcount + 1;   // + NUM_BG
}

extern "C" void kernel_launch(void* const* d_in, const int* in_sizes, int n_in,
                              void* d_out, int out_size, void* d_ws, size_t ws_size,
                              hipStream_t stream) {
    (void)in_sizes; (void)n_in; (void)out_size; (void)ws_size;
    const float* sim = (const float*)d_in[0];
    // d_in[1] = category_ids (unused by the reference's forward path)
    const int* ori = (const int*)d_in[2];

    u64* ws = (u64*)d_ws;                               // 8192 cells * 2 u64 = 128 KB
    float* out_pts = (float*)d_out;                     // (T,C,42,4) = 5376 floats
    int* out_num = (int*)d_out + (size_t)T_ * C_ * MAX_PTS * 4;  // 32 int32 (bit-stored)

    cell_reduce_kernel<<<dim3(T_ * C_ * NUM_GRID * NUM_GRID), 256, 0, stream>>>(sim, ws);
    select_kernel<<<dim3(T_ * C_), 256, 0, stream>>>(ws, ori, out_pts, out_num);
}